// SpatialAttentionModule_16578573762818
// MI455X (gfx1250) — compile-verified
//
#include <hip/hip_runtime.h>
#include <math.h>

// Problem constants
#define B_TOTAL 32
#define GROUP   8            // 8 images * 12.8MB = 102MB < 192MB L2 -> pass-3 re-read hits L2
#define C       256
#define H       112
#define W       112
#define HW      (H * W)      // 12544
#define CHW     (C * HW)     // 3211264

#define CSLICES 4            // channel-split factor for pass 1
#define CPER    (C / CSLICES)    // 64 channels per thread
#define SPB     64               // spatial positions per block (HW % 64 == 0)

typedef __attribute__((ext_vector_type(2))) float v2f;
typedef __attribute__((ext_vector_type(4))) float v4f;
typedef __attribute__((ext_vector_type(8))) float v8f;

// ---------------------------------------------------------------------------
// Pass 1: channel-wise max + mean, channel-split 4-ways for latency hiding.
// Block = 256 threads = 64 spatial positions x 4 channel slices of 64 ch.
// Each wave = 32 consecutive spatial addrs in one slice -> 128B coalesced.
// ~12.5K waves with unroll-16 keeps ~26MB of loads in flight (BW*latency).
// Partials combined through LDS. Leaves x resident in L2 for pass 3.
// ---------------------------------------------------------------------------
__global__ void pool_kernel(const float* __restrict__ x, float* __restrict__ pooled) {
    __shared__ float smax[CSLICES * SPB];
    __shared__ float ssum[CSLICES * SPB];

    int tx     = threadIdx.x;
    int cslice = tx >> 6;            // 0..3
    int sIdx   = tx & 63;            // 0..63
    unsigned p = blockIdx.x * SPB + sIdx;   // global spatial position in group
    unsigned b = p / HW;             // whole block is within one image
    unsigned s = p - b * HW;

    const float* ptr = x + (size_t)b * CHW + (size_t)(cslice * CPER) * HW + s;
    float mx  = -3.402823466e38f;
    float sum = 0.0f;
#pragma unroll 16
    for (int c = 0; c < CPER; ++c) {
        float v = ptr[(size_t)c * HW];
        mx  = fmaxf(mx, v);
        sum += v;
    }
    smax[tx] = mx;
    ssum[tx] = sum;
    __syncthreads();

    if (tx < SPB) {
        float m0 = fmaxf(fmaxf(smax[tx], smax[SPB + tx]),
                         fmaxf(smax[2 * SPB + tx], smax[3 * SPB + tx]));
        float s0 = (ssum[tx] + ssum[SPB + tx]) +
                   (ssum[2 * SPB + tx] + ssum[3 * SPB + tx]);
        pooled[(size_t)b * (2 * HW) + s]      = m0;
        pooled[(size_t)b * (2 * HW) + HW + s] = s0 * (1.0f / C);
    }
}

// ---------------------------------------------------------------------------
// Pass 2: 7x7 conv (2 ch) + bias + sigmoid via V_WMMA_F32_16X16X4_F32.
// Each wave computes 16 output pixels (M = pixels). K = 98 taps padded to 100,
// processed as 25 chunks of 4. B holds the weight chunk broadcast across all
// 16 columns, so every column of D equals the conv result.
// f32 A layout (ISA 7.12.2): lane m=lane&15; VGPR0 = K=(0|2) by lane-half,
// VGPR1 = K=(1|3).  B mirrors this in K.
// ---------------------------------------------------------------------------
__device__ __forceinline__ float patchval(const float* __restrict__ pb,
                                          int py, int px, int k) {
    int ci = k / 49;
    int t  = k - ci * 49;
    int kh = t / 7;
    int kw = t - kh * 7;
    int iy = py + kh - 3;
    int ix = px + kw - 3;
    float v = 0.0f;
    if ((unsigned)iy < (unsigned)H && (unsigned)ix < (unsigned)W)
        v = pb[ci * HW + iy * W + ix];
    return v;
}

__global__ void conv_wmma_kernel(const float* __restrict__ pooled,
                                 const float* __restrict__ wgt,
                                 const float* __restrict__ bias,
                                 float* __restrict__ mask) {
    __shared__ float sw[104];                    // 98 weights + zero padding
    int tid = threadIdx.x;
    if (tid < 104) sw[tid] = (tid < 98) ? wgt[tid] : 0.0f;
    __syncthreads();

    int lane = tid & 31;
    int wave = blockIdx.x * (blockDim.x >> 5) + (tid >> 5);
    int b    = wave / (HW / 16);                 // image within group
    int tile = wave - b * (HW / 16);
    int s0   = tile * 16;                        // 16 consecutive flat pixels
    int half = lane >> 4;
    int m    = lane & 15;
    int s    = s0 + m;
    int py   = s / W;
    int px   = s - py * W;
    const float* pb = pooled + (size_t)b * (2 * HW);

    v8f acc = {0.f, 0.f, 0.f, 0.f, 0.f, 0.f, 0.f, 0.f};
#pragma unroll 5
    for (int kc = 0; kc < 25; ++kc) {
        int k0  = kc * 4 + 2 * half;             // this lane-half's K pair
        int ka0 = (k0     < 97) ? k0       : 97; // clamp A reads in-bounds;
        int ka1 = (k0 + 1 < 97) ? (k0 + 1) : 97; // padded taps have w == 0
        v2f a, bv;
        a.x  = patchval(pb, py, px, ka0);
        a.y  = patchval(pb, py, px, ka1);
        bv.x = sw[k0];
        bv.y = sw[k0 + 1];
        acc = __builtin_amdgcn_wmma_f32_16x16x4_f32(
                  false, a, false, bv, (short)0, acc, false, false);
    }

    // D layout: vgpr j, low half -> M=j ; high half -> M=j+8. All columns
    // identical, so lane 0 / lane 16 hold the 16 results.
    if (m == 0) {
        float bi = bias[0];
        float* mp = mask + (size_t)b * HW + s0 + 8 * half;
#pragma unroll
        for (int j = 0; j < 8; ++j) {
            float y = acc[j] + bi;
            mp[j] = 1.0f / (1.0f + __expf(-y));
        }
    }
}

// ---------------------------------------------------------------------------
// Pass 3: out = mask * x. float4 streaming; x re-read should hit L2 (group
// fits). Non-temporal load/store keep the dead output stream from evicting
// the mask / x lines (TH=NT on gfx1250).
// ---------------------------------------------------------------------------
__global__ void apply_kernel(const float* __restrict__ x,
                             const float* __restrict__ mask,
                             float* __restrict__ out) {
    size_t i4 = (size_t)blockIdx.x * blockDim.x + threadIdx.x; // GROUP*CHW/4
    size_t e  = i4 * 4;
    unsigned b = (unsigned)(e / CHW);
    unsigned r = (unsigned)(e - (size_t)b * CHW);
    unsigned s = r % HW;                        // e%4==0 -> s%4==0, aligned
    v4f xv = __builtin_nontemporal_load((const v4f*)(x + e));
    v4f mv = *(const v4f*)(mask + (size_t)b * HW + s);
    v4f ov = xv * mv;
    __builtin_nontemporal_store(ov, (v4f*)(out + e));
}

// ---------------------------------------------------------------------------
extern "C" void kernel_launch(void* const* d_in, const int* in_sizes, int n_in,
                              void* d_out, int out_size, void* d_ws, size_t ws_size,
                              hipStream_t stream) {
    (void)in_sizes; (void)n_in; (void)out_size; (void)ws_size;
    const float* x    = (const float*)d_in[0];
    const float* wgt  = (const float*)d_in[1];   // 1*2*7*7 = 98 floats
    const float* bias = (const float*)d_in[2];   // 1 float
    float* out = (float*)d_out;

    // Workspace (stream-serialized, reused per group):
    float* pooled = (float*)d_ws;                       // GROUP*2*HW floats
    float* mask   = pooled + (size_t)GROUP * 2 * HW;    // GROUP*HW floats

    for (int g = 0; g < B_TOTAL; g += GROUP) {
        const float* xg = x   + (size_t)g * CHW;
        float*       og = out + (size_t)g * CHW;

        int poolBlocks = GROUP * HW / SPB;                      // 1568 blocks
        pool_kernel<<<dim3(poolBlocks), dim3(256), 0, stream>>>(xg, pooled);

        int waves = GROUP * (HW / 16);                          // 6272 waves
        conv_wmma_kernel<<<dim3(waves / 8), dim3(256), 0, stream>>>(pooled, wgt, bias, mask);

        size_t applyThreads = (size_t)GROUP * CHW / 4;          // 6,422,528
        apply_kernel<<<dim3((unsigned)(applyThreads / 256)), dim3(256), 0, stream>>>(xg, mask, og);
    }
}